// TopKRouter_19739669692844
// MI455X (gfx1250) — compile-verified
//
#include <hip/hip_runtime.h>

#define NTOK   131072
#define DIM    1024
#define NEXP   64
#define KCHUNK 16
#define NSTAGE (DIM / KCHUNK)   // 64
#define XPAD   20               // row stride in floats: 80 B = 16B-aligned, gcd(20,64)=4 -> conflict-free frags
#define WPAD   20
#define LGPAD  65               // 64 experts + 1 pad
#define TPB    128              // tokens per block (8 waves x 16)

typedef float v2f __attribute__((ext_vector_type(2)));
typedef float v4f __attribute__((ext_vector_type(4)));
typedef float v8f __attribute__((ext_vector_type(8)));

#if __has_builtin(__builtin_amdgcn_global_load_async_to_lds_b128)
#define USE_ASYNC 1
#else
#define USE_ASYNC 0
#endif

#if USE_ASYNC
// Builtin signature (from compiler diagnostic): pointers to gcc-vector <4 x i32>
typedef int v4i __attribute__((vector_size(16)));
typedef __attribute__((address_space(1))) v4i* gv4_t;
typedef __attribute__((address_space(3))) v4i* lv4_t;

__device__ __forceinline__ void async_copy16(const float* g, float* l) {
    __builtin_amdgcn_global_load_async_to_lds_b128(
        (gv4_t)(uintptr_t)g, (lv4_t)(uintptr_t)l, 0, 0);
}

__device__ __forceinline__ void wait_async0() {
#if __has_builtin(__builtin_amdgcn_s_wait_asynccnt)
    __builtin_amdgcn_s_wait_asynccnt(0);
#else
    asm volatile("s_wait_asynccnt 0" ::: "memory");
#endif
}
#endif

__global__ __launch_bounds__(128) void router_zero(float* g) {
    g[threadIdx.x] = 0.0f;   // zeros probsum[64] + usage[64]
}

__global__ __launch_bounds__(256) void router_main(
    const float* __restrict__ x, const float* __restrict__ gw,
    float* __restrict__ outw, float* __restrict__ outi,
    float* __restrict__ gprob, float* __restrict__ gusage)
{
    __shared__ float xs[2][TPB * XPAD];   // 2 x 128x20 floats = 20480 B
    __shared__ float wl[2][NEXP * WPAD];  // 2 x  64x20 floats = 10240 B
    __shared__ float lg[TPB * LGPAD];     // 128x65 floats     = 33280 B
    __shared__ float ps[NEXP];
    __shared__ float us[NEXP];

    const int tid  = threadIdx.x;
    const int wave = tid >> 5;
    const int lane = tid & 31;
    const int half = lane >> 4;          // K sub-offset selector
    const int lrow = lane & 15;          // M (for A) / N (for B) within tile
    const size_t tokBase = (size_t)blockIdx.x * TPB;
    const float* xsrc = x + tokBase * DIM;

    // Per-thread b128 staging slots: x chunk = 128 rows x 16 floats, w chunk = 64 rows x 16 floats
    const int xrow0 = tid >> 2;          // rows   0..63
    const int xrow1 = (tid + 256) >> 2;  // rows  64..127
    const int skp   = (tid & 3) * 4;     // float4 column within the 16-float chunk
    const int wrow  = tid >> 2;          // rows   0..63

    if (tid < NEXP) { ps[tid] = 0.0f; us[tid] = 0.0f; }

    // ---- stage 0 fill ----
#if USE_ASYNC
    async_copy16(xsrc + (size_t)xrow0 * DIM + skp, &xs[0][xrow0 * XPAD + skp]);
    async_copy16(xsrc + (size_t)xrow1 * DIM + skp, &xs[0][xrow1 * XPAD + skp]);
    async_copy16(gw   + (size_t)wrow  * DIM + skp, &wl[0][wrow  * WPAD + skp]);
    wait_async0();
#else
    {
        v4f a0 = *(const v4f*)(xsrc + (size_t)xrow0 * DIM + skp);
        v4f a1 = *(const v4f*)(xsrc + (size_t)xrow1 * DIM + skp);
        v4f w0 = *(const v4f*)(gw   + (size_t)wrow  * DIM + skp);
        *(v4f*)(&xs[0][xrow0 * XPAD + skp]) = a0;
        *(v4f*)(&xs[0][xrow1 * XPAD + skp]) = a1;
        *(v4f*)(&wl[0][wrow  * WPAD + skp]) = w0;
    }
#endif
    __syncthreads();

    v8f acc[4] = {};   // 4 expert tiles of 16x16 f32

    for (int s = 0; s < NSTAGE; ++s) {
        const int cur = s & 1, nxt = cur ^ 1;

        // prefetch 2 stages ahead into caches (global_prefetch_b8)
        if (s + 2 < NSTAGE) {
            __builtin_prefetch(xsrc + (size_t)xrow0 * DIM + (s + 2) * KCHUNK + skp, 0, 1);
        }

#if USE_ASYNC
        // DMA next stage straight into the other LDS buffer while WMMAs run
        if (s + 1 < NSTAGE) {
            const int kb = (s + 1) * KCHUNK;
            async_copy16(xsrc + (size_t)xrow0 * DIM + kb + skp, &xs[nxt][xrow0 * XPAD + skp]);
            async_copy16(xsrc + (size_t)xrow1 * DIM + kb + skp, &xs[nxt][xrow1 * XPAD + skp]);
            async_copy16(gw   + (size_t)wrow  * DIM + kb + skp, &wl[nxt][wrow  * WPAD + skp]);
        }
#else
        v4f rx0, rx1, rw0;
        if (s + 1 < NSTAGE) {
            const int kb = (s + 1) * KCHUNK;
            rx0 = *(const v4f*)(xsrc + (size_t)xrow0 * DIM + kb + skp);
            rx1 = *(const v4f*)(xsrc + (size_t)xrow1 * DIM + kb + skp);
            rw0 = *(const v4f*)(gw   + (size_t)wrow  * DIM + kb + skp);
        }
#endif

        // ---- WMMA compute on current stage ----
        #pragma unroll
        for (int kk = 0; kk < KCHUNK; kk += 4) {
            // A fragment: lane(lrow,half) -> x[wave*16+lrow][kk + 2*half .. +1]
            v2f a = *(const v2f*)(&xs[cur][(wave * 16 + lrow) * XPAD + kk + half * 2]);
            #pragma unroll
            for (int t = 0; t < 4; ++t) {
                // B fragment: lane(lrow,half) -> gate_w[t*16+lrow][kk + 2*half .. +1]
                v2f b = *(const v2f*)(&wl[cur][(t * 16 + lrow) * WPAD + kk + half * 2]);
                acc[t] = __builtin_amdgcn_wmma_f32_16x16x4_f32(
                    false, a, false, b, (short)0, acc[t], false, false);
            }
        }

#if USE_ASYNC
        wait_async0();                 // this wave's DMA into nxt has landed
#else
        if (s + 1 < NSTAGE) {
            *(v4f*)(&xs[nxt][xrow0 * XPAD + skp]) = rx0;
            *(v4f*)(&xs[nxt][xrow1 * XPAD + skp]) = rx1;
            *(v4f*)(&wl[nxt][wrow  * WPAD + skp]) = rw0;
        }
#endif
        __syncthreads();
    }

    // ---- spill logits: C layout -> lg[token][expert] ----
    #pragma unroll
    for (int r = 0; r < 8; ++r) {
        int tokLocal = wave * 16 + half * 8 + r;   // M = r + 8*half
        #pragma unroll
        for (int t = 0; t < 4; ++t) {
            lg[tokLocal * LGPAD + t * 16 + lrow] = acc[t][r];
        }
    }
    __syncthreads();

    // ---- fused softmax / top-2 / aux reductions: one token per thread ----
    if (tid < TPB) {
        const float* lr = &lg[tid * LGPAD];
        float m1 = -3.4e38f; int i1 = 0;
        #pragma unroll 8
        for (int e = 0; e < NEXP; ++e) { float l = lr[e]; if (l > m1) { m1 = l; i1 = e; } }
        float m2 = -3.4e38f; int i2 = (i1 == 0) ? 1 : 0;
        #pragma unroll 8
        for (int e = 0; e < NEXP; ++e) { if (e == i1) continue; float l = lr[e]; if (l > m2) { m2 = l; i2 = e; } }
        float ssum = 0.0f;
        #pragma unroll 8
        for (int e = 0; e < NEXP; ++e) ssum += __expf(lr[e] - m1);
        float inv = 1.0f / ssum;
        #pragma unroll 8
        for (int e = 0; e < NEXP; ++e) atomicAdd(&ps[e], __expf(lr[e] - m1) * inv);
        atomicAdd(&us[i1], 1.0f);
        atomicAdd(&us[i2], 1.0f);

        // normalized top-2 weights: p1/(p1+p2) = 1/(1+exp(l2-l1))
        float w1 = 1.0f / (1.0f + __expf(m2 - m1));
        size_t g = tokBase + tid;
        outw[2 * g]     = w1;
        outw[2 * g + 1] = 1.0f - w1;
        outi[2 * g]     = (float)i1;
        outi[2 * g + 1] = (float)i2;
    }
    __syncthreads();

    if (tid < NEXP) {
        atomicAdd(&gprob[tid],  ps[tid]);
        atomicAdd(&gusage[tid], us[tid]);
    }
}

__global__ __launch_bounds__(64) void router_fin(
    const float* __restrict__ gprob, const float* __restrict__ gusage,
    float* __restrict__ aux)
{
    __shared__ float red[NEXP];
    int t = threadIdx.x;
    red[t] = gprob[t] * gusage[t];
    __syncthreads();
    for (int off = 32; off > 0; off >>= 1) {
        if (t < off) red[t] += red[t + off];
        __syncthreads();
    }
    if (t == 0) {
        float nn = (float)NTOK;
        aux[0] = (float)NEXP * red[0] / (nn * nn);
    }
}

extern "C" void kernel_launch(void* const* d_in, const int* in_sizes, int n_in,
                              void* d_out, int out_size, void* d_ws, size_t ws_size,
                              hipStream_t stream) {
    const float* x  = (const float*)d_in[0];   // [131072, 1024]
    const float* gw = (const float*)d_in[1];   // [64, 1024]

    float* out  = (float*)d_out;
    float* outw = out;                          // [N,2] normalized top weights
    float* outi = out + 2 * (size_t)NTOK;       // [N,2] indices (as float)
    float* aux  = out + 4 * (size_t)NTOK;       // [1]

    float* gprob  = (float*)d_ws;               // [64]
    float* gusage = gprob + NEXP;               // [64]

    router_zero<<<1, 128, 0, stream>>>(gprob);
    router_main<<<NTOK / TPB, 256, 0, stream>>>(x, gw, outw, outi, gprob, gusage);
    router_fin<<<1, 64, 0, stream>>>(gprob, gusage, aux);
}